// WindAwareAdjacency_56788057588038
// MI455X (gfx1250) — compile-verified
//
#include <hip/hip_runtime.h>
#include <math.h>

typedef float v2f __attribute__((ext_vector_type(2)));
typedef float v8f __attribute__((ext_vector_type(8)));

__device__ __forceinline__ float sigmoid_precise(float x) {
    return 1.0f / (1.0f + expf(-x));
}
__device__ __forceinline__ float sigmoid_fast(float x) {
    // v_exp_f32 + v_rcp_f32 — cheap transcendental path for the hot loop
    return __builtin_amdgcn_rcpf(1.0f + __expf(-x));
}

// ---------------------------------------------------------------------------
// Kernel 1 (one block, 32 waves): fold all (b,t)-invariant terms into
//   base = fixed_adj + softmax(relu(E1 @ E2^T), axis=1)
//          + sig(gamma)*mean(sig(kappa)) * D^-1/2 * fixed_adj * D^-1/2
// GEMM done with V_WMMA_F32_16X16X4_F32 (full fp32), relu(S) staged in LDS.
// ---------------------------------------------------------------------------
__global__ __launch_bounds__(1024, 1) void precompute_base_kernel(
    const float* __restrict__ E1, const float* __restrict__ E2,
    const float* __restrict__ gamma_p, const float* __restrict__ kappa,
    const float* __restrict__ fixed_adj, float* __restrict__ base,
    int E, int F)
{
    __shared__ float S[128 * 128];     // 64 KB of the 320 KB WGP LDS
    __shared__ float dinv_s[128];

    const int tid  = threadIdx.x;
    const int lane = tid & 31;
    const int wave = tid >> 5;
    const int m    = lane & 15;        // M (for A) / N (for B) index
    const int hi   = lane >> 4;        // half-wave select: K += 2*hi

    // ---- S = relu(E1 @ E2^T) via WMMA f32 16x16x4, 8x8 tiles of 16x16 ----
    for (int t = wave; t < 64; t += 32) {          // uniform per wave: EXEC all-1s
        const int ti = t >> 3, tj = t & 7;
        const float* arow = E1 + (ti * 16 + m) * E;   // A: row-major E1
        const float* brow = E2 + (tj * 16 + m) * E;   // B[k][n] = E2[n][k]
        v8f c = {0.f, 0.f, 0.f, 0.f, 0.f, 0.f, 0.f, 0.f};
        for (int kk = 0; kk < E; kk += 4) {
            const int kb = kk + 2 * hi;
            v2f a = { arow[kb], arow[kb + 1] };       // K = kb, kb+1
            v2f b = { brow[kb], brow[kb + 1] };
            c = __builtin_amdgcn_wmma_f32_16x16x4_f32(
                    false, a, false, b, (short)0, c, false, false);
        }
#pragma unroll
        for (int v = 0; v < 8; ++v) {
            // C/D layout: row = v + 8*hi, col = m
            S[(ti * 16 + v + 8 * hi) * 128 + (tj * 16 + m)] = fmaxf(c[v], 0.0f);
        }
    }

    // ---- degrees -> d_inv_sqrt ----
    for (int r = wave; r < 128; r += 32) {
        float s = 0.0f;
#pragma unroll
        for (int cdx = 0; cdx < 4; ++cdx) s += fixed_adj[r * 128 + lane + cdx * 32];
#pragma unroll
        for (int off = 16; off >= 1; off >>= 1) s += __shfl_xor(s, off, 32);
        if (lane == 0) dinv_s[r] = 1.0f / (sqrtf(s) + 1e-8f);
    }
    __syncthreads();

    // kappa_mean * sig(gamma) (tiny, computed redundantly)
    float km = 0.0f;
    for (int f = 0; f < F; ++f) km += sigmoid_precise(kappa[f]);
    km /= (float)F;
    const float cdiff = sigmoid_precise(gamma_p[0]) * km;

    // ---- per-row softmax + assemble base ----
    for (int r = wave; r < 128; r += 32) {
        const int j0 = lane * 4;
        float4 s = *(const float4*)&S[r * 128 + j0];
        float mx = fmaxf(fmaxf(s.x, s.y), fmaxf(s.z, s.w));
#pragma unroll
        for (int off = 16; off >= 1; off >>= 1) mx = fmaxf(mx, __shfl_xor(mx, off, 32));
        float4 e;
        e.x = expf(s.x - mx); e.y = expf(s.y - mx);
        e.z = expf(s.z - mx); e.w = expf(s.w - mx);
        float sm = e.x + e.y + e.z + e.w;
#pragma unroll
        for (int off = 16; off >= 1; off >>= 1) sm += __shfl_xor(sm, off, 32);
        const float inv = 1.0f / sm;
        const float4 fa = *(const float4*)&fixed_adj[r * 128 + j0];
        const float dr = dinv_s[r] * cdiff;
        float4 o;
        o.x = fa.x + e.x * inv + dr * dinv_s[j0 + 0] * fa.x;
        o.y = fa.y + e.y * inv + dr * dinv_s[j0 + 1] * fa.y;
        o.z = fa.z + e.z * inv + dr * dinv_s[j0 + 2] * fa.z;
        o.w = fa.w + e.w * inv + dr * dinv_s[j0 + 3] * fa.w;
        *(float4*)&base[r * 128 + j0] = o;
    }
}

// ---------------------------------------------------------------------------
// Kernel 2 (B*T blocks x 256 threads = 8 waves): per (b,t) slice.
// Wave owns row i; each lane handles 4 j's with b128 loads/stores.
// Row-normalization via wave32 xor-shuffle reduction.
// ---------------------------------------------------------------------------
__global__ __launch_bounds__(256, 4) void wind_adjacency_kernel(
    const float* __restrict__ wind_u, const float* __restrict__ wind_v,
    const float* __restrict__ pressure,
    const float* __restrict__ alpha_p, const float* __restrict__ beta_p,
    const float* __restrict__ base, float* __restrict__ out,
    const float* __restrict__ dir_lon, const float* __restrict__ dir_lat,
    const float* __restrict__ inv_dist)
{
    __shared__ __align__(16) float wu_s[128];
    __shared__ __align__(16) float wv_s[128];
    __shared__ __align__(16) float p_s[128];

    const int bt   = blockIdx.x;
    const int tid  = threadIdx.x;
    const int lane = tid & 31;
    const int wave = tid >> 5;

    if (tid < 128) {
        wu_s[tid] = wind_u[bt * 128 + tid];
        wv_s[tid] = wind_v[bt * 128 + tid];
        p_s[tid]  = pressure[bt * 128 + tid];
    }
    __syncthreads();

    const float sa = sigmoid_precise(alpha_p[0]);
    const float sb = sigmoid_precise(beta_p[0]);

    const int j0 = lane * 4;
    const float4 wuj = *(const float4*)&wu_s[j0];
    const float4 wvj = *(const float4*)&wv_s[j0];
    const float4 pj  = *(const float4*)&p_s[j0];

    float* outbt = out + (size_t)bt * 16384;

    for (int i = wave; i < 128; i += 8) {
        const float wui = wu_s[i], wvi = wv_s[i], pi = p_s[i];
        const float4 dl = *(const float4*)&dir_lon[i * 128 + j0];
        const float4 dt = *(const float4*)&dir_lat[i * 128 + j0];
        const float4 iv = *(const float4*)&inv_dist[i * 128 + j0];
        const float4 bs = *(const float4*)&base[i * 128 + j0];

        float4 t;
        t.x = fmaxf((wui - wuj.x) * dl.x + (wvi - wvj.x) * dt.x, 0.0f) * 0.5f
              * sigmoid_fast(sb * (pi - pj.x)) * iv.x;
        t.y = fmaxf((wui - wuj.y) * dl.y + (wvi - wvj.y) * dt.y, 0.0f) * 0.5f
              * sigmoid_fast(sb * (pi - pj.y)) * iv.y;
        t.z = fmaxf((wui - wuj.z) * dl.z + (wvi - wvj.z) * dt.z, 0.0f) * 0.5f
              * sigmoid_fast(sb * (pi - pj.z)) * iv.z;
        t.w = fmaxf((wui - wuj.w) * dl.w + (wvi - wvj.w) * dt.w, 0.0f) * 0.5f
              * sigmoid_fast(sb * (pi - pj.w)) * iv.w;

        float sum = t.x + t.y + t.z + t.w;
#pragma unroll
        for (int off = 16; off >= 1; off >>= 1) sum += __shfl_xor(sum, off, 32);
        const float scale = sa / (sum + 1e-8f);

        float4 o;
        o.x = bs.x + t.x * scale;
        o.y = bs.y + t.y * scale;
        o.z = bs.z + t.z * scale;
        o.w = bs.w + t.w * scale;
        *(float4*)&outbt[i * 128 + j0] = o;
    }
}

extern "C" void kernel_launch(void* const* d_in, const int* in_sizes, int n_in,
                              void* d_out, int out_size, void* d_ws, size_t ws_size,
                              hipStream_t stream) {
    const float* wind_u    = (const float*)d_in[0];
    const float* wind_v    = (const float*)d_in[1];
    const float* pressure  = (const float*)d_in[2];
    const float* E1        = (const float*)d_in[3];
    const float* E2        = (const float*)d_in[4];
    const float* alpha_p   = (const float*)d_in[5];
    const float* beta_p    = (const float*)d_in[6];
    const float* gamma_p   = (const float*)d_in[7];
    const float* kappa     = (const float*)d_in[8];
    const float* fixed_adj = (const float*)d_in[9];
    const float* dir_lon   = (const float*)d_in[10];
    const float* dir_lat   = (const float*)d_in[11];
    const float* inv_dist  = (const float*)d_in[12];

    float* out  = (float*)d_out;
    float* base = (float*)d_ws;              // N*N fp32 = 64 KB scratch

    const int N  = 128;
    const int E  = in_sizes[3] / N;          // 64
    const int F  = in_sizes[8];              // 18
    const int BT = in_sizes[0] / N;          // 1536

    precompute_base_kernel<<<1, 1024, 0, stream>>>(
        E1, E2, gamma_p, kappa, fixed_adj, base, E, F);
    wind_adjacency_kernel<<<BT, 256, 0, stream>>>(
        wind_u, wind_v, pressure, alpha_p, beta_p, base, out,
        dir_lon, dir_lat, inv_dist);
}